// rep_gcc_Conv2d_79224966742716
// MI455X (gfx1250) — compile-verified
//
#include <hip/hip_runtime.h>
#include <hip/hip_bf16.h>

// out[b,c,h,w] = sum_k G_c[h,k] * (x[b,c,k,w] + pe[c,k]) + bias[c]
// G_c[h,k] = weight[c,(k-h)%H] + (rep_weight[c,(k-h)%H] if (k-h)%H < 3)
//
// Batched GEMM per channel: Out(96x1536) = G_c(96x96) * X(96x1536) + beta,
// with beta[c,h] = bias[c] + sum_k G_c[h,k]*pe[c,k] folded into the epilogue
// (pe is n-independent, so it never belongs in the hot loop).
// V_WMMA_F32_16X16X4_F32: fp32-exact; compute hidden under the ~13us HBM
// floor (0.3 GB at 23.3 TB/s). Each wave owns a 96(h) x 16(n) strip so x is
// read exactly once (NT loads) and out written exactly once (NT stores).
// All 8 waves of a block share one channel, so the circulant A fragments
// (full 36 KB G_c in WMMA register layout) and beta are built once per
// block in LDS; hot loop = 2 NT loads + 6 ds_load_b64 + 6 WMMAs.

typedef float v2f __attribute__((ext_vector_type(2)));
typedef float v8f __attribute__((ext_vector_type(8)));

#define BATCH 16
#define CDIM  256
#define HDIM  96
#define WDIM  96
#define REPK  3

#define NT_PER_IMG (WDIM / 16)            // 6 n-tiles per image row-block
#define NT         (BATCH * NT_PER_IMG)   // 96 n-tiles per channel
#define MTILES     (HDIM / 16)            // 6 h-tiles per strip
#define KSTEPS     (HDIM / 4)             // 24 K-steps of 4
#define WAVES_PER_BLOCK 8
#define AFRAG_N    (KSTEPS * MTILES * 32) // 4608 float2 = 36,864 B LDS

__global__ __launch_bounds__(256)
void circ_dwconv_wmma_f32(const float* __restrict__ x,
                          const float* __restrict__ pe,
                          const float* __restrict__ weight,
                          const float* __restrict__ bias,
                          const float* __restrict__ repw,
                          float* __restrict__ out)
{
    __shared__ float2 afrag[AFRAG_N];     // G_c in WMMA A-fragment layout
    __shared__ float  betab[HDIM];        // bias[c] + (G_c x pe_c)[h]

    const int lane = threadIdx.x & 31;
    const int wid  = blockIdx.x * WAVES_PER_BLOCK + (threadIdx.x >> 5);

    const int c  = wid / NT;              // uniform across the block (96%8==0)
    const int nt = wid % NT;
    const int bb = nt / NT_PER_IMG;       // batch index
    const int w0 = (nt % NT_PER_IMG) * 16;

    const float* wc = weight + c * HDIM;  // 384 B, cache-resident
    const float* rc = repw   + c * REPK;
    const float* pc = pe     + c * HDIM;
    const float* xc = x   + (size_t)(bb * CDIM + c) * HDIM * WDIM + w0;
    float*       oc = out + (size_t)(bb * CDIM + c) * HDIM * WDIM + w0;

    // ---- build A fragments (whole 96x96 circulant) in LDS, once per block ----
    // entry e -> (kkidx, mt, lane): afrag[e] = (G[h][kx], G[h][kx+1]) with
    // h = 16*mt + (lane&15), kx = 4*kkidx + 2*(lane>>4)  (WMMA 16x4 f32 layout)
    for (int e = threadIdx.x; e < AFRAG_N; e += 256) {   // 4608/256: no remainder
        const int lane_e = e & 31;
        const int mt_e   = (e >> 5) % MTILES;
        const int kk_e   = e / (32 * MTILES);
        const int kx = kk_e * 4 + ((lane_e >> 4) * 2);
        const int h  = mt_e * 16 + (lane_e & 15);
        int j0 = kx - h; if (j0 < 0) j0 += HDIM;
        int j1 = j0 + 1; if (j1 >= HDIM) j1 -= HDIM;
        const float g0 = wc[j0] + ((j0 < REPK) ? rc[j0] : 0.0f);
        const float g1 = wc[j1] + ((j1 < REPK) ? rc[j1] : 0.0f);
        afrag[e] = make_float2(g0, g1);
    }

    // ---- beta[h] = bias[c] + sum_k G_c[h,k] * pe[c,k]  (once per block) ----
    if (threadIdx.x < HDIM) {
        const int h = threadIdx.x;
        float acc = bias[c];
        for (int k = 0; k < HDIM; ++k) {
            int j = k - h; if (j < 0) j += HDIM;
            const float g = wc[j] + ((j < REPK) ? rc[j] : 0.0f);
            acc += g * pc[k];
        }
        betab[h] = acc;
    }
    __syncthreads();

    const int m  = lane & 15;             // A row (M) / B,C column (N) in tile
    const int kb = (lane >> 4) * 2;       // K sub-offset for this half-wave

    v8f acc[MTILES] = {};                 // 6 x 16x16 f32 accumulators

    for (int kkidx = 0; kkidx < KSTEPS; ++kkidx) {
        const int kx = kkidx * 4 + kb;    // this half-wave covers rows kx, kx+1

        // ---- B tile: x rows are the only HBM stream -> non-temporal loads ----
        v2f b;
        b.x = __builtin_nontemporal_load(&xc[kx * WDIM + m]);
        b.y = __builtin_nontemporal_load(&xc[(kx + 1) * WDIM + m]);

#pragma unroll
        for (int mt = 0; mt < MTILES; ++mt) {
            // conflict-free ds_load_b64: lane-consecutive float2 spans 64 banks
            const float2 af = afrag[(kkidx * MTILES + mt) * 32 + lane];
            v2f a; a.x = af.x; a.y = af.y;

            acc[mt] = __builtin_amdgcn_wmma_f32_16x16x4_f32(
                /*neg_a=*/false, a, /*neg_b=*/false, b,
                /*c_mod=*/(short)0, acc[mt],
                /*reuse_a=*/false, /*reuse_b=*/false);
        }
    }

    // ---- epilogue: + beta[h], NT store of the 96x16 strip (written once) ----
    const int rowadd = (lane >> 4) * 8;   // upper half-wave holds rows M=8..15
#pragma unroll
    for (int mt = 0; mt < MTILES; ++mt) {
#pragma unroll
        for (int j = 0; j < 8; ++j) {
            const int h = mt * 16 + j + rowadd;
            __builtin_nontemporal_store(acc[mt][j] + betab[h], &oc[h * WDIM + m]);
        }
    }
}

extern "C" void kernel_launch(void* const* d_in, const int* in_sizes, int n_in,
                              void* d_out, int out_size, void* d_ws, size_t ws_size,
                              hipStream_t stream) {
    const float* x      = (const float*)d_in[0];  // (16,256,96,96)
    const float* pe     = (const float*)d_in[1];  // (1,256,96,1)
    const float* weight = (const float*)d_in[2];  // (256,1,96,1)
    const float* bias   = (const float*)d_in[3];  // (256,)
    const float* repw   = (const float*)d_in[4];  // (256,1,3,1)
    float* out = (float*)d_out;                   // (16,256,96,96)

    const int total_waves = CDIM * NT;                       // 24576
    const int blocks = total_waves / WAVES_PER_BLOCK;        // 3072
    circ_dwconv_wmma_f32<<<blocks, 256, 0, stream>>>(x, pe, weight, bias, repw, out);
}